// Transducer_5669356832197
// MI455X (gfx1250) — compile-verified
//
#include <hip/hip_runtime.h>
#include <cstddef>
#include <cstdint>

// Problem constants (from reference)
#define B_   8
#define T_   500
#define U_   120
#define U1_  121
#define V_   128
#define H_   320
#define G_   1280   // 4*H
#define DIN_ 80

#define T_P   560   // padded per-batch row stride for EE (7 * 80)
#define U1_P  128   // padded per-batch row stride for DE
#define HD_P  1040  // padded rows for HD (13 * 80)

typedef float v2f __attribute__((ext_vector_type(2)));
typedef float v8f __attribute__((ext_vector_type(8)));
typedef unsigned int u32x4 __attribute__((ext_vector_type(4)));
typedef int i32x4 __attribute__((ext_vector_type(4)));
typedef int i32x8 __attribute__((ext_vector_type(8)));

// D = A(16x4 f32) * B(4x16 f32) + C on the CDNA5 tensor pipe (full f32).
__device__ __forceinline__ v8f wmma_f32_16x16x4(v2f a, v2f b, v8f c) {
  return __builtin_amdgcn_wmma_f32_16x16x4_f32(false, a, false, b, (short)0, c,
                                               false, false);
}

__device__ __forceinline__ v2f ld2(const float2* p, int i) {
  float2 t = p[i];
  v2f r;
  r.x = t.x;
  r.y = t.y;
  return r;
}

// ---------------------------------------------------------------------------
// Tensor Data Mover: async 2-D tile (nRows x rowElems f32) -> LDS.
// D# per CDNA5 ISA ss8.3/8.4: group0 = {count=1, lds_addr,
// global_addr[56:0], type=2}; group1 = {data_size=4B, tensor_dim0/1,
// tile_dim0/1, tensor_dim0_stride}. Groups 2/3 zero (2-D tensor).
// ---------------------------------------------------------------------------
__device__ __forceinline__ void tdm_load_rows_to_lds(const float* gsrc,
                                                     unsigned lds_byte,
                                                     int rowElems, int nRows,
                                                     int rowStrideElems) {
  unsigned long long ga = (unsigned long long)(uintptr_t)gsrc;
  u32x4 g0;
  g0.x = 1u;                                            // count=1 (valid D#)
  g0.y = lds_byte;                                      // lds_addr
  g0.z = (unsigned)(ga & 0xFFFFFFFFull);                // global_addr[31:0]
  g0.w = (unsigned)((ga >> 32) & 0x1FFFFFFull) |        // global_addr[56:32]
         (2u << 30);                                    // type=2 ("image")
  i32x8 g1;
  g1[0] = 2 << 16;                                      // data_size=4B
  g1[1] = (rowElems & 0xFFFF) << 16;                    // tensor_dim0[15:0]
  g1[2] = ((rowElems >> 16) & 0xFFFF) |                 // tensor_dim0[31:16]
          ((nRows & 0xFFFF) << 16);                     // tensor_dim1[15:0]
  g1[3] = (rowElems & 0xFFFF) << 16;                    // tile_dim0
  g1[4] = nRows & 0xFFFF;                               // tile_dim1 (dim2=0)
  g1[5] = rowStrideElems;                               // dim0_stride[31:0]
  g1[6] = 0;
  g1[7] = 0;
  i32x4 gz = {0, 0, 0, 0};
#if __clang_major__ >= 23
  i32x8 gz8 = {0, 0, 0, 0, 0, 0, 0, 0};
  __builtin_amdgcn_tensor_load_to_lds(g0, g1, gz, gz, gz8, 0);
#else
  __builtin_amdgcn_tensor_load_to_lds(g0, g1, gz, gz, 0);
#endif
}

// ---------------------------------------------------------------------------
// Y[M,N] = X[M,K] @ W[N,K]^T + bias0 + bias1.
// One wave computes an 80x32 macro-tile (5 M-tiles x 2 N-tiles). K-loop is
// unrolled by two 4-deep WMMA steps with disjoint fragment registers so both
// sub-steps' loads are in flight before the first WMMA group issues.
// Caller guarantees X rows up to gridDim.x*80 readable (padded workspace);
// stores masked by M.
// ---------------------------------------------------------------------------
__global__ void gemm80x32_wmma(const float* __restrict__ X,
                               const float* __restrict__ W,
                               const float* __restrict__ bias0,
                               const float* __restrict__ bias1,
                               float* __restrict__ Y, int M, int N, int K) {
  const int lane = threadIdx.x;
  const int lr = lane & 15;
  const int kh = lane >> 4;
  const int row0 = blockIdx.x * 80;
  const int col0 = blockIdx.y * 32;
  const float2* xr[5];
  const float2* wr[2];
#pragma unroll
  for (int i = 0; i < 5; ++i)
    xr[i] = (const float2*)(X + (size_t)(row0 + i * 16 + lr) * K);
#pragma unroll
  for (int j = 0; j < 2; ++j)
    wr[j] = (const float2*)(W + (size_t)(col0 + j * 16 + lr) * K);
  v8f acc[5][2] = {};
  for (int k0 = 0; k0 < K; k0 += 8) {      // two 4-deep WMMA steps per iter
    const int ki0 = (k0 >> 1) + kh;
    const int ki1 = ki0 + 2;
    v2f a0[5], a1[5], b0[2], b1[2];
#pragma unroll
    for (int i = 0; i < 5; ++i) a0[i] = ld2(xr[i], ki0);
#pragma unroll
    for (int j = 0; j < 2; ++j) b0[j] = ld2(wr[j], ki0);
#pragma unroll
    for (int i = 0; i < 5; ++i) a1[i] = ld2(xr[i], ki1);
#pragma unroll
    for (int j = 0; j < 2; ++j) b1[j] = ld2(wr[j], ki1);
#pragma unroll
    for (int i = 0; i < 5; ++i)
#pragma unroll
      for (int j = 0; j < 2; ++j)
        acc[i][j] = wmma_f32_16x16x4(a0[i], b0[j], acc[i][j]);
#pragma unroll
    for (int i = 0; i < 5; ++i)
#pragma unroll
      for (int j = 0; j < 2; ++j)
        acc[i][j] = wmma_f32_16x16x4(a1[i], b1[j], acc[i][j]);
  }
#pragma unroll
  for (int j = 0; j < 2; ++j) {
    const int n = col0 + j * 16 + lr;
    float bn = 0.f;
    if (bias0) bn += bias0[n];
    if (bias1) bn += bias1[n];
#pragma unroll
    for (int i = 0; i < 5; ++i) {
#pragma unroll
      for (int r = 0; r < 8; ++r) {
        int mm = row0 + i * 16 + r + (kh << 3);
        if (mm < M) Y[(size_t)mm * N + n] = acc[i][j][r] + bn;
      }
    }
  }
}

// ---------------------------------------------------------------------------
// Persistent LSTM scan: one workgroup (16 waves). Per step:
//   g = xproj[:,t,:] + h @ Whh^T  (WMMA f32 16x16x4, batch padded 8->16)
// xproj rows for step t+1 are staged into double-buffered LDS by the TDM
// while step t computes; h/c/gates live in LDS.
// ---------------------------------------------------------------------------
__global__ void lstm_scan_wmma(const float* __restrict__ xproj,
                               const float* __restrict__ Whh,
                               float* __restrict__ hs, int T) {
  extern __shared__ float smem[];
  float* h_lds = smem;                  // 16 x H (rows 8..15 stay zero)
  float* c_lds = h_lds + 16 * H_;       // 8 x H
  float* g_lds = c_lds + 8 * H_;        // 8 x G
  float* xp0   = g_lds + 8 * G_;        // 8 x G  (TDM buffer 0)
  float* xp1   = xp0 + 8 * G_;          // 8 x G  (TDM buffer 1)
  float* xp[2] = {xp0, xp1};
  const unsigned xpb[2] = {(unsigned)(uintptr_t)xp0, (unsigned)(uintptr_t)xp1};
  const int tid = threadIdx.x;
  const int lane = tid & 31;
  const int wave = tid >> 5;            // 16 waves
  const int lr = lane & 15;
  const int kh = lane >> 4;

  for (int i = tid; i < 16 * H_; i += blockDim.x) h_lds[i] = 0.f;
  for (int i = tid; i < 8 * H_; i += blockDim.x) c_lds[i] = 0.f;

  // Stage xproj row-block for t=0 (8 rows of G, batch stride T*G).
  if (tid < 32) {
    tdm_load_rows_to_lds(xproj, xpb[0], G_, 8, T * G_);
    __builtin_amdgcn_s_wait_tensorcnt((short)0);
  }
  __syncthreads();

  const float2* hr = (const float2*)(h_lds + lr * H_);
  const float2* wr[5];
#pragma unroll
  for (int nt = 0; nt < 5; ++nt)
    wr[nt] = (const float2*)(Whh + (size_t)((wave * 5 + nt) * 16 + lr) * H_);

  int cur = 0;
  for (int t = 0; t < T; ++t) {
    // Kick off async staging of step t+1's xproj block into the other buffer.
    if (tid < 32 && (t + 1) < T)
      tdm_load_rows_to_lds(xproj + (size_t)(t + 1) * G_, xpb[cur ^ 1], G_, 8,
                           T * G_);

    // 80 gate column tiles: 16 waves x 5 tiles; A-frag (h) shared over tiles.
    v8f acc[5] = {};
    for (int k0 = 0; k0 < H_; k0 += 8) {   // two 4-deep WMMA steps per iter
      const int ki0 = (k0 >> 1) + kh;
      const int ki1 = ki0 + 2;
      v2f a0 = ld2(hr, ki0);
      v2f a1 = ld2(hr, ki1);
      v2f b0[5], b1[5];
#pragma unroll
      for (int nt = 0; nt < 5; ++nt) b0[nt] = ld2(wr[nt], ki0);
#pragma unroll
      for (int nt = 0; nt < 5; ++nt) b1[nt] = ld2(wr[nt], ki1);
#pragma unroll
      for (int nt = 0; nt < 5; ++nt)
        acc[nt] = wmma_f32_16x16x4(a0, b0[nt], acc[nt]);
#pragma unroll
      for (int nt = 0; nt < 5; ++nt)
        acc[nt] = wmma_f32_16x16x4(a1, b1[nt], acc[nt]);
    }
    if (kh == 0) {                      // rows 0..7 = real batch rows
#pragma unroll
      for (int nt = 0; nt < 5; ++nt) {
        const int n = (wave * 5 + nt) * 16 + lr;
#pragma unroll
        for (int r = 0; r < 8; ++r) g_lds[r * G_ + n] = acc[nt][r];
      }
    }
    __syncthreads();

    // Elementwise gate math + state update; xproj read from LDS (TDM-staged).
    const float* xr = xp[cur];
    for (int idx = tid; idx < 8 * H_; idx += blockDim.x) {
      int b = idx / H_;
      int j = idx - b * H_;
      float gi = g_lds[b * G_ + j]          + xr[b * G_ + j];
      float gf = g_lds[b * G_ + H_ + j]     + xr[b * G_ + H_ + j];
      float gc = g_lds[b * G_ + 2 * H_ + j] + xr[b * G_ + 2 * H_ + j];
      float go = g_lds[b * G_ + 3 * H_ + j] + xr[b * G_ + 3 * H_ + j];
      float i_ = 1.f / (1.f + expf(-gi));
      float f_ = 1.f / (1.f + expf(-gf));
      float o_ = 1.f / (1.f + expf(-go));
      float g_ = tanhf(gc);
      float c = f_ * c_lds[idx] + i_ * g_;
      c_lds[idx] = c;
      float hn = o_ * tanhf(c);
      h_lds[b * H_ + j] = hn;
      hs[((size_t)b * T + t) * H_ + j] = hn;
    }
    if (tid < 32) __builtin_amdgcn_s_wait_tensorcnt((short)0);
    __syncthreads();
    cur ^= 1;
  }
}

// ---------------------------------------------------------------------------
// Decoder input projection: embed is [0 ; I], so x@Wih^T is a column gather.
// ---------------------------------------------------------------------------
__global__ void dec_inproj(const int* __restrict__ ys,
                           const float* __restrict__ Wih,
                           const float* __restrict__ bih,
                           const float* __restrict__ bhh,
                           float* __restrict__ xprojD) {
  int idx = blockIdx.x * blockDim.x + threadIdx.x;
  const int total = B_ * U1_ * G_;
  if (idx >= total) return;
  int g = idx % G_;
  int bu = idx / G_;
  int u = bu % U1_;
  int b = bu / U1_;
  float v = bih[g] + bhh[g];
  if (u > 0) {
    int y = ys[b * U_ + (u - 1)];            // in [1, V-1]
    v += Wih[(size_t)g * (V_ - 1) + (y - 1)];
  }
  xprojD[idx] = v;
}

// ---------------------------------------------------------------------------
// Row max + exp(x - max); output rows re-strided to a padded per-batch pitch
// so the joint GEMM can load unmasked.
// ---------------------------------------------------------------------------
__global__ void rowmax_exp(const float* __restrict__ X, float* __restrict__ mrow,
                           float* __restrict__ E, int rowsPerB, int rowPitch) {
  __shared__ float red[V_];
  int r = blockIdx.x;
  int t = threadIdx.x;
  float v = X[(size_t)r * V_ + t];
  red[t] = v;
  __syncthreads();
  for (int s = V_ / 2; s > 0; s >>= 1) {
    if (t < s) red[t] = fmaxf(red[t], red[t + s]);
    __syncthreads();
  }
  float m = red[0];
  size_t orow = (size_t)(r / rowsPerB) * rowPitch + (r % rowsPerB);
  E[orow * V_ + t] = expf(v - m);
  if (t == 0) mrow[r] = m;
}

// ---------------------------------------------------------------------------
// Joint without materializing [B,T,U+1,V]:
//   S = Eexp[T,128] @ Dexp[U+1,128]^T  (WMMA f32, K=128, 80x32 macro-tile)
//   lse = log(S)+me+md;  lb = enc0+dec0-lse;  ly = enc[y]+dec[y]-lse
// ---------------------------------------------------------------------------
__global__ void joint_lse_wmma(const float* __restrict__ EE,
                               const float* __restrict__ DE,
                               const float* __restrict__ me,
                               const float* __restrict__ md,
                               const float* __restrict__ enc,
                               const float* __restrict__ dec,
                               const int* __restrict__ ys,
                               float* __restrict__ lb, float* __restrict__ ly) {
  const int lane = threadIdx.x;
  const int lr = lane & 15;
  const int kh = lane >> 4;
  const int b = blockIdx.z;
  const int t0 = blockIdx.x * 80;
  const int u0 = blockIdx.y * 32;
  const float2* er[5];
  const float2* dr[2];
#pragma unroll
  for (int i = 0; i < 5; ++i)
    er[i] = (const float2*)(EE + ((size_t)b * T_P + t0 + i * 16 + lr) * V_);
#pragma unroll
  for (int j = 0; j < 2; ++j)
    dr[j] = (const float2*)(DE + ((size_t)b * U1_P + u0 + j * 16 + lr) * V_);
  v8f acc[5][2] = {};
  for (int k0 = 0; k0 < V_; k0 += 8) {     // two 4-deep WMMA steps per iter
    const int ki0 = (k0 >> 1) + kh;
    const int ki1 = ki0 + 2;
    v2f a0[5], a1[5], b0[2], b1[2];
#pragma unroll
    for (int i = 0; i < 5; ++i) a0[i] = ld2(er[i], ki0);
#pragma unroll
    for (int j = 0; j < 2; ++j) b0[j] = ld2(dr[j], ki0);
#pragma unroll
    for (int i = 0; i < 5; ++i) a1[i] = ld2(er[i], ki1);
#pragma unroll
    for (int j = 0; j < 2; ++j) b1[j] = ld2(dr[j], ki1);
#pragma unroll
    for (int i = 0; i < 5; ++i)
#pragma unroll
      for (int j = 0; j < 2; ++j)
        acc[i][j] = wmma_f32_16x16x4(a0[i], b0[j], acc[i][j]);
#pragma unroll
    for (int i = 0; i < 5; ++i)
#pragma unroll
      for (int j = 0; j < 2; ++j)
        acc[i][j] = wmma_f32_16x16x4(a1[i], b1[j], acc[i][j]);
  }
#pragma unroll
  for (int i = 0; i < 5; ++i) {
#pragma unroll
    for (int j = 0; j < 2; ++j) {
      const int uu = u0 + j * 16 + lr;
#pragma unroll
      for (int r = 0; r < 8; ++r) {
        const int tt = t0 + i * 16 + r + (kh << 3);
        if (tt < T_ && uu < U1_) {
          float lse = logf(acc[i][j][r]) + me[b * T_ + tt] + md[b * U1_ + uu];
          const float* e2 = enc + ((size_t)b * T_ + tt) * V_;
          const float* d2 = dec + ((size_t)b * U1_ + uu) * V_;
          size_t o = ((size_t)b * T_ + tt) * U1_ + uu;
          lb[o] = e2[0] + d2[0] - lse;
          if (uu < U_) {
            int y = ys[b * U_ + uu];
            ly[o] = e2[y] + d2[y] - lse;
          }
        }
      }
    }
  }
}

// ---------------------------------------------------------------------------
// Alpha recursion (reference closed form): per t row,
//   a = alpha_prev + lb[t-1];  c = prefix-sum(ly[t]);
//   alpha = c + cumlogsumexp(a - c)
// 128-lane Hillis-Steele scans in LDS; one workgroup per sample.
// ---------------------------------------------------------------------------
__device__ __forceinline__ float logaddexp_(float x, float y) {
  float m = fmaxf(x, y);
  return m + log1pf(expf(fminf(x, y) - m));
}

__global__ void rnnt_alpha(const float* __restrict__ lb,
                           const float* __restrict__ ly,
                           const int* __restrict__ xlen,
                           const int* __restrict__ ylen,
                           float* __restrict__ nll) {
  __shared__ float s_alpha[128];
  __shared__ float s_c[128];
  __shared__ float s_w[128];
  const int b = blockIdx.x;
  const int u = threadIdx.x;
  const float* LB = lb + (size_t)b * T_ * U1_;
  const float* LY = ly + (size_t)b * T_ * U1_;
  const int tl = xlen[b];
  const int ul = ylen[b];
  const float NEG = -3.0e38f;

  s_c[u] = (u >= 1 && u <= U_) ? LY[u - 1] : 0.f;
  __syncthreads();
  for (int off = 1; off < 128; off <<= 1) {
    float v = s_c[u];
    float w = (u >= off) ? s_c[u - off] : 0.f;
    __syncthreads();
    s_c[u] = v + w;
    __syncthreads();
  }
  s_alpha[u] = (u < U1_) ? s_c[u] : NEG;
  __syncthreads();

  for (int t = 1; t < tl; ++t) {
    float a = (u < U1_) ? s_alpha[u] + LB[(size_t)(t - 1) * U1_ + u] : NEG;
    s_c[u] = (u >= 1 && u <= U_) ? LY[(size_t)t * U1_ + (u - 1)] : 0.f;
    __syncthreads();
    for (int off = 1; off < 128; off <<= 1) {
      float v = s_c[u];
      float w = (u >= off) ? s_c[u - off] : 0.f;
      __syncthreads();
      s_c[u] = v + w;
      __syncthreads();
    }
    s_w[u] = (u < U1_) ? a - s_c[u] : NEG;
    __syncthreads();
    for (int off = 1; off < 128; off <<= 1) {
      float v = s_w[u];
      float w = (u >= off) ? s_w[u - off] : NEG;
      __syncthreads();
      s_w[u] = logaddexp_(v, w);
      __syncthreads();
    }
    s_alpha[u] = (u < U1_) ? s_c[u] + s_w[u] : NEG;
    __syncthreads();
  }
  if (u == 0) nll[b] = -(s_alpha[ul] + LB[(size_t)(tl - 1) * U1_ + ul]);
}

__global__ void mean_nll(const float* __restrict__ nll, float* __restrict__ out) {
  if (threadIdx.x == 0) {
    float s = 0.f;
    for (int i = 0; i < B_; ++i) s += nll[i];
    out[0] = s / (float)B_;
  }
}

// ---------------------------------------------------------------------------
extern "C" void kernel_launch(void* const* d_in, const int* in_sizes, int n_in,
                              void* d_out, int out_size, void* d_ws,
                              size_t ws_size, hipStream_t stream) {
  (void)in_sizes; (void)n_in; (void)out_size; (void)ws_size;
  const float* xs   = (const float*)d_in[0];
  const int*   ys   = (const int*)d_in[1];
  const int*   xlen = (const int*)d_in[2];
  const int*   ylen = (const int*)d_in[3];
  const float* Wih0 = (const float*)d_in[4];
  const float* Whh0 = (const float*)d_in[5];
  const float* bih0 = (const float*)d_in[6];
  const float* bhh0 = (const float*)d_in[7];
  const float* Wih1 = (const float*)d_in[8];
  const float* Whh1 = (const float*)d_in[9];
  const float* bih1 = (const float*)d_in[10];
  const float* bhh1 = (const float*)d_in[11];
  const float* Wo   = (const float*)d_in[12];
  const float* bo   = (const float*)d_in[13];
  /* d_in[14] = embed: unused, structure exploited analytically */
  const float* dWih = (const float*)d_in[15];
  const float* dWhh = (const float*)d_in[16];
  const float* dbih = (const float*)d_in[17];
  const float* dbhh = (const float*)d_in[18];
  const float* dWo  = (const float*)d_in[19];
  const float* dbo  = (const float*)d_in[20];
  float* out = (float*)d_out;

  float* ws = (float*)d_ws;
  size_t off = 0;
  auto alloc = [&](size_t n) {
    float* p = ws + off;
    off += (n + 63) & ~(size_t)63;
    return p;
  };
  const int BT = B_ * T_;    // 4000
  const int BU = B_ * U1_;   // 968
  float* P   = alloc((size_t)BT * G_);        // xproj0 then xproj1 (reused)
  float* HS0 = alloc((size_t)BT * H_);
  float* HS1 = alloc((size_t)BT * H_);
  float* ENC = alloc((size_t)BT * V_);
  float* PD  = alloc((size_t)BU * G_);
  float* HD  = alloc((size_t)HD_P * H_);      // padded: dec GEMM loads unmasked
  float* DEC = alloc((size_t)BU * V_);
  float* ME  = alloc(BT);
  float* MD  = alloc(BU);
  float* EE  = alloc((size_t)B_ * T_P * V_);  // padded per-batch pitch
  float* DE  = alloc((size_t)B_ * U1_P * V_); // padded per-batch pitch
  float* LB  = alloc((size_t)BT * U1_);
  float* LY  = alloc((size_t)BT * U1_);
  float* NLL = alloc(64);

  const size_t smem = (size_t)(16 * H_ + 8 * H_ + 3 * 8 * G_) * sizeof(float);

  // Encoder layer 0
  gemm80x32_wmma<<<dim3(BT / 80, G_ / 32), 32, 0, stream>>>(xs, Wih0, bih0,
                                                            bhh0, P, BT, G_,
                                                            DIN_);
  lstm_scan_wmma<<<1, 512, smem, stream>>>(P, Whh0, HS0, T_);
  // Encoder layer 1
  gemm80x32_wmma<<<dim3(BT / 80, G_ / 32), 32, 0, stream>>>(HS0, Wih1, bih1,
                                                            bhh1, P, BT, G_,
                                                            H_);
  lstm_scan_wmma<<<1, 512, smem, stream>>>(P, Whh1, HS1, T_);
  // Encoder projection
  gemm80x32_wmma<<<dim3(BT / 80, V_ / 32), 32, 0, stream>>>(HS1, Wo, bo,
                                                            nullptr, ENC, BT,
                                                            V_, H_);
  // Decoder
  dec_inproj<<<(B_ * U1_ * G_ + 255) / 256, 256, 0, stream>>>(ys, dWih, dbih,
                                                              dbhh, PD);
  lstm_scan_wmma<<<1, 512, smem, stream>>>(PD, dWhh, HD, U1_);
  gemm80x32_wmma<<<dim3(HD_P / 80, V_ / 32), 32, 0, stream>>>(HD, dWo, dbo,
                                                              nullptr, DEC, BU,
                                                              V_, H_);
  // Factored log-softmax joint
  rowmax_exp<<<BT, V_, 0, stream>>>(ENC, ME, EE, T_, T_P);
  rowmax_exp<<<BU, V_, 0, stream>>>(DEC, MD, DE, U1_, U1_P);
  joint_lse_wmma<<<dim3((T_ + 79) / 80, (U1_ + 31) / 32, B_), 32, 0, stream>>>(
      EE, DE, ME, MD, ENC, DEC, ys, LB, LY);
  // Alpha recursion + mean
  rnnt_alpha<<<B_, 128, 0, stream>>>(LB, LY, xlen, ylen, NLL);
  mean_nll<<<1, 32, 0, stream>>>(NLL, out);
}